// Attention_37890201486072
// MI455X (gfx1250) — compile-verified
//
#include <hip/hip_runtime.h>

// ---------------------------------------------------------------------------
// Attention-pooling (scores -> softmax over T -> weighted sum), MI455X/gfx1250.
// B=32, T=2048, H=512, fp32. x is 128 MiB -> bandwidth bound (23.3 TB/s floor
// ~5.5us/pass); x fits in 192MB L2 so pass 2 re-reads from L2.
// K3 uses V_WMMA_F32_16X16X4_F32 for out = X^T * w (coalesced A loads).
// ---------------------------------------------------------------------------

typedef __attribute__((ext_vector_type(2))) float v2f;
typedef __attribute__((ext_vector_type(8))) float v8f;

#define BB 32
#define TT 2048
#define HH 512
#define ROWS_PER_WAVE 64                       // K1: rows per wave
#define WAVES_PER_BATCH (TT / ROWS_PER_WAVE)   // 32
#define NCHUNK (BB * WAVES_PER_BATCH)          // 1024 partial maxes

__device__ __forceinline__ float dot4(float4 a, float4 b, float p) {
    p = fmaf(a.x, b.x, p);
    p = fmaf(a.y, b.y, p);
    p = fmaf(a.z, b.z, p);
    p = fmaf(a.w, b.w, p);
    return p;
}

// K1: scores[b,t] = x[b,t,:].W + bias ; per-wave-chunk max partials.
// 128 blocks x 256 thr = 1024 waves; wave w -> batch w/32, rows (w%32)*64..+63.
__global__ __launch_bounds__(256) void k_scores(
        const float* __restrict__ X, const float* __restrict__ Wv,
        const float* __restrict__ bias,
        float* __restrict__ scores, float* __restrict__ chunkmax) {
    const int lane   = threadIdx.x & 31;
    const int waveId = blockIdx.x * (blockDim.x >> 5) + (threadIdx.x >> 5);
    const int b      = waveId >> 5;       // /WAVES_PER_BATCH
    const int t0     = (waveId & 31) * ROWS_PER_WAVE;

    // Preload W (512 f32) into registers: 4 float4 per lane.
    const float4* W4 = (const float4*)Wv;
    float4 w0 = W4[lane], w1 = W4[lane + 32], w2 = W4[lane + 64], w3 = W4[lane + 96];
    const float bs = bias[0];

    const float4* Xb = (const float4*)(X + (size_t)b * TT * HH);
    float mloc = -3.402823466e38f;

    for (int r = 0; r < ROWS_PER_WAVE; ++r) {
        const float4* row = Xb + (size_t)(t0 + r) * (HH / 4);
        __builtin_prefetch(row + 4 * (HH / 4) + lane, 0, 0);  // 4 rows ahead
        float4 x0 = row[lane], x1 = row[lane + 32];
        float4 x2 = row[lane + 64], x3 = row[lane + 96];
        float p = 0.0f;
        p = dot4(x0, w0, p); p = dot4(x1, w1, p);
        p = dot4(x2, w2, p); p = dot4(x3, w3, p);
        #pragma unroll
        for (int off = 16; off >= 1; off >>= 1) p += __shfl_xor(p, off, 32);
        p += bs;
        if (lane == 0) scores[b * TT + t0 + r] = p;
        mloc = fmaxf(mloc, p);
    }
    if (lane == 0) chunkmax[waveId] = mloc;
}

// K2: per batch: m = max(chunkmax), e=exp(s-m), S=sum(e), weights=e/S.
// Writes weights directly into d_out's attention_weights region.
__global__ __launch_bounds__(256) void k_softmax(
        const float* __restrict__ scores, const float* __restrict__ chunkmax,
        float* __restrict__ wout) {
    const int b = blockIdx.x, tid = threadIdx.x;
    __shared__ float sm, sinv, red[8];

    if (tid < 32) {
        float m = chunkmax[b * 32 + tid];
        #pragma unroll
        for (int off = 16; off >= 1; off >>= 1) m = fmaxf(m, __shfl_xor(m, off, 32));
        if (tid == 0) sm = m;
    }
    __syncthreads();
    const float m = sm;

    float lsum = 0.0f;
    for (int t = tid; t < TT; t += 256) {
        float e = expf(scores[b * TT + t] - m);
        wout[b * TT + t] = e;                 // unnormalized for now
        lsum += e;
    }
    #pragma unroll
    for (int off = 16; off >= 1; off >>= 1) lsum += __shfl_xor(lsum, off, 32);
    if ((tid & 31) == 0) red[tid >> 5] = lsum;
    __syncthreads();
    if (tid == 0) {
        float s = 0.0f;
        #pragma unroll
        for (int i = 0; i < 8; ++i) s += red[i];
        sinv = 1.0f / s;
    }
    __syncthreads();
    const float inv = sinv;
    for (int t = tid; t < TT; t += 256) wout[b * TT + t] *= inv;
}

// K3: out[b,h] = sum_t w[b,t] * x[b,t,h]  via V_WMMA_F32_16X16X4_F32.
// Grid (4 h-groups, 32 batches), 256 thr = 8 waves; wave -> 16-wide h tile.
// A[m][k] = x[t0+k][h0+m]  (A layout: lanes0-15 K=0/1 in v0/v1, lanes16-31 K=2/3
//  -> per A register, two 64B coalesced row segments).
// B[k][n] = w[t0+k] broadcast over n (B layout: lanes0-15 K=0/1, lanes16-31 K=2/3).
// D accumulates over all T; every N column identical -> lanes 0/16 write out.
__global__ __launch_bounds__(256) void k_weighted_sum(
        const float* __restrict__ X, const float* __restrict__ wts,
        float* __restrict__ out) {
    const int b    = blockIdx.y;
    const int lane = threadIdx.x & 31;
    const int wv   = threadIdx.x >> 5;
    const int h0   = blockIdx.x * 128 + wv * 16;

    __shared__ float lw[TT];                  // 8 KB staged weights row
    for (int t = threadIdx.x; t < TT; t += 256) lw[t] = wts[b * TT + t];
    __syncthreads();

    const float* Xb = X + (size_t)b * TT * HH;
    const int ro = (lane >> 4) << 1;          // 0 for lanes 0-15, 2 for 16-31
    const int hc = h0 + (lane & 15);

    v8f acc = {};
    #pragma unroll 4
    for (int t0 = 0; t0 < TT; t0 += 4) {
        __builtin_prefetch(Xb + (size_t)(t0 + 8 + ro) * HH + hc, 0, 0);
        v2f a;
        a[0] = Xb[(size_t)(t0 + ro)     * HH + hc];
        a[1] = Xb[(size_t)(t0 + ro + 1) * HH + hc];
        v2f bw = *(const v2f*)&lw[t0 + ro];   // ds_load_b64, 8B aligned
        acc = __builtin_amdgcn_wmma_f32_16x16x4_f32(
            /*neg_a=*/false, a, /*neg_b=*/false, bw,
            /*c_mod=*/(short)0, acc, /*reuse_a=*/false, /*reuse_b=*/false);
    }

    // D[m][n]: VGPR r: lanes0-15 -> M=r (N=lane); lanes16-31 -> M=8+r.
    if (lane == 0) {
        #pragma unroll
        for (int r = 0; r < 8; ++r) out[b * HH + h0 + r] = acc[r];
    } else if (lane == 16) {
        #pragma unroll
        for (int r = 0; r < 8; ++r) out[b * HH + h0 + 8 + r] = acc[r];
    }
}

extern "C" void kernel_launch(void* const* d_in, const int* in_sizes, int n_in,
                              void* d_out, int out_size, void* d_ws, size_t ws_size,
                              hipStream_t stream) {
    const float* X    = (const float*)d_in[0];   // [32,2048,512]
    const float* Wv   = (const float*)d_in[1];   // [512,1]
    const float* bias = (const float*)d_in[2];   // [1]

    float* out  = (float*)d_out;                 // [32,512]
    float* wout = out + BB * HH;                 // [32,2048] attention weights

    float* scores   = (float*)d_ws;              // B*T floats
    float* chunkmax = scores + BB * TT;          // NCHUNK floats

    k_scores<<<dim3(NCHUNK / 8), dim3(256), 0, stream>>>(X, Wv, bias, scores, chunkmax);
    k_softmax<<<dim3(BB), dim3(256), 0, stream>>>(scores, chunkmax, wout);
    k_weighted_sum<<<dim3(HH / 128, BB), dim3(256), 0, stream>>>(X, wout, out);
}